// GlacierDynamicsCheckpointed_41412074668209
// MI455X (gfx1250) — compile-verified
//
#include <hip/hip_runtime.h>
#include <hip/hip_bf16.h>
#include <stdint.h>

// ---------------- problem constants ----------------
#define NXY   512
#define NTOT  (512 * 512)
#define NDAYS 365
#define NSTEPS 64

typedef __attribute__((ext_vector_type(2))) float v2f;
typedef __attribute__((ext_vector_type(8))) float v8f;

// FD * (RHO*G)^3 ; RHO*G = 910*9.81 = 8927.1
__device__ const float CD_CONST = (float)(3.1e-17 * 8927.1 * 8927.1 * 8927.1);

// =====================================================================
// k_init: H_init -> Hbuf0, zero snapshot slices, zero per-step maxD,
//         initialize scalar slot 0 and the "initial SMB" slot (index NSTEPS).
// =====================================================================
__global__ void k_init(const float* __restrict__ H0, float* __restrict__ Hb0,
                       float* __restrict__ out, unsigned* __restrict__ maxDbits,
                       float* ts, float* tl, int* pyi,
                       int* g26, int* g57, int* g80, int* needA, int* yiA) {
  int idx = blockIdx.x * 256 + threadIdx.x;
  Hb0[idx] = H0[idx];
  out[idx] = 0.0f;                 // s26
  out[NTOT + idx] = 0.0f;          // s57
  out[2 * NTOT + idx] = 0.0f;      // s80
  if (idx < NSTEPS) maxDbits[idx] = 0u;
  if (idx == 0) {
    ts[0] = 1979.0f;               // TIME0
    tl[0] = 0.0f;                  // t_last
    pyi[0] = 115;                  // floor(1979-1864)
    g26[0] = 0; g57[0] = 0; g80[0] = 0;
    needA[NSTEPS] = 1;             // initial SMB: unconditional
    yiA[NSTEPS] = 115;
  }
}

// =====================================================================
// k_diff: D = CD * H^5 * |grad(Zt+H)|^2 (jnp.gradient edge rules),
//         wave32 shuffle + LDS block max, atomicMax into per-step slot.
// =====================================================================
__global__ void k_diff(const float* __restrict__ Zt, const float* __restrict__ H,
                       float* __restrict__ D, unsigned* __restrict__ maxDbits) {
  int idx = blockIdx.x * 256 + threadIdx.x;
  int x = idx & (NXY - 1), y = idx >> 9;
  float Hc = H[idx];
  float Sc = Zt[idx] + Hc;
  float Sxp = (x < NXY - 1) ? Zt[idx + 1] + H[idx + 1] : Sc;
  float Sxm = (x > 0)       ? Zt[idx - 1] + H[idx - 1] : Sc;
  float Syp = (y < NXY - 1) ? Zt[idx + NXY] + H[idx + NXY] : Sc;
  float Sym = (y > 0)       ? Zt[idx - NXY] + H[idx - NXY] : Sc;
  float gx = (x == 0) ? (Sxp - Sc) * 0.01f
           : (x == NXY - 1) ? (Sc - Sxm) * 0.01f
           : (Sxp - Sxm) * 0.005f;
  float gy = (y == 0) ? (Syp - Sc) * 0.01f
           : (y == NXY - 1) ? (Sc - Sym) * 0.01f
           : (Syp - Sym) * 0.005f;
  float H2 = Hc * Hc;
  float Dv = CD_CONST * (H2 * H2 * Hc) * (gx * gx + gy * gy);
  D[idx] = Dv;

  // block max (wave32 reduce, then 8-wave LDS reduce)
  float m = Dv;
  #pragma unroll
  for (int off = 16; off > 0; off >>= 1) m = fmaxf(m, __shfl_down(m, off, 32));
  __shared__ float red[8];
  int lane = threadIdx.x & 31, w = threadIdx.x >> 5;
  if (lane == 0) red[w] = m;
  __syncthreads();
  if (threadIdx.x == 0) {
    float mm = red[0];
    #pragma unroll
    for (int i = 1; i < 8; ++i) mm = fmaxf(mm, red[i]);
    atomicMax(maxDbits, __float_as_uint(mm));  // D >= 0: uint compare == float compare
  }
}

// =====================================================================
// k_update: dt from per-step maxD, flux divergence, H update, snapshots,
//           and single-thread scalar advance (slot step -> step+1).
// =====================================================================
__global__ void k_update(const float* __restrict__ Zt, const float* __restrict__ Hin,
                         float* __restrict__ Hout, const float* __restrict__ D,
                         const float* __restrict__ smb, const unsigned* __restrict__ maxDbits,
                         float* ts, float* tl, int* pyi,
                         int* g26, int* g57, int* g80, int* needA, int* yiA,
                         int step, float* __restrict__ out) {
  int idx = blockIdx.x * 256 + threadIdx.x;
  float t = ts[step];
  float Dmax = __uint_as_float(maxDbits[step]);
  float dt = fminf(0.1f, 2000.0f / (Dmax + 1e-6f));  // CFL*min(DX,DY)^2 = 0.2*1e4
  bool active = t < 1981.0f;                          // TTOT
  float Hc = Hin[idx];
  float Hn = Hc;
  float tn = t;
  if (active) {
    tn = t + dt;
    int x = idx & (NXY - 1), y = idx >> 9;
    float Sc = Zt[idx] + Hc;
    float Dc = D[idx];
    float fe = 0.f, fw = 0.f, fn = 0.f, fs = 0.f;
    if (x < NXY - 1) { float S1 = Zt[idx + 1]   + Hin[idx + 1];   fe = 0.5f * (Dc + D[idx + 1])   * (S1 - Sc) * 0.01f; }
    if (x > 0)       { float S1 = Zt[idx - 1]   + Hin[idx - 1];   fw = 0.5f * (Dc + D[idx - 1])   * (Sc - S1) * 0.01f; }
    if (y < NXY - 1) { float S1 = Zt[idx + NXY] + Hin[idx + NXY]; fn = 0.5f * (Dc + D[idx + NXY]) * (S1 - Sc) * 0.01f; }
    if (y > 0)       { float S1 = Zt[idx - NXY] + Hin[idx - NXY]; fs = 0.5f * (Dc + D[idx - NXY]) * (Sc - S1) * 0.01f; }
    float div = (fe - fw) * 0.01f + (fn - fs) * 0.01f;
    Hn = fmaxf(Hc + dt * (smb[idx] + div), 0.0f);
    // snapshots: hit = active & (tn >= yr); write when hit & ~g
    if (tn >= 1926.0f && !g26[step]) out[idx] = Hn;
    if (tn >= 1957.0f && !g57[step]) out[NTOT + idx] = Hn;
    if (tn >= 1980.0f && !g80[step]) out[2 * NTOT + idx] = Hn;
  }
  Hout[idx] = Hn;

  if (idx == 0) {  // race-free: everyone reads slot `step`, only this writes `step+1`
    int a = active ? 1 : 0;
    g26[step + 1] = g26[step] | (a & (tn >= 1926.0f ? 1 : 0));
    g57[step + 1] = g57[step] | (a & (tn >= 1957.0f ? 1 : 0));
    g80[step + 1] = g80[step] | (a & (tn >= 1980.0f ? 1 : 0));
    ts[step + 1] = tn;
    int yi = (int)floorf(tn - 1864.0f);
    yi = yi < 0 ? 0 : (yi > 127 ? 127 : yi);
    int need = a && ((yi != pyi[step]) || (tn - tl[step] >= 10.0f));
    needA[step] = need;
    yiA[step] = yi;
    tl[step + 1] = need ? tn : tl[step];
    pyi[step + 1] = need ? yi : pyi[step];
  }
}

// branchless 8-way select (avoids dynamic vector index -> scratch)
__device__ __forceinline__ float mux8(const v8f& v, int c) {
  float m0 = (c & 1) ? v[1] : v[0];
  float m1 = (c & 1) ? v[3] : v[2];
  float m2 = (c & 1) ? v[5] : v[4];
  float m3 = (c & 1) ? v[7] : v[6];
  float n0 = (c & 2) ? m1 : m0;
  float n1 = (c & 2) ? m3 : m2;
  return (c & 4) ? n1 : n0;
}

// =====================================================================
// k_smb: conditional degree-day SMB. Day rows staged into LDS with
// CDNA5 async-to-LDS (ASYNCcnt); the 365-day pdd/acc accumulation runs
// on the matrix pipe via V_WMMA_F32_16X16X4_F32 with B = ones (row-sum).
// LDS tails padded (T=-1e30 -> relu 0, P=0 -> snow 0) so the inner loop
// is straight-line: ds_load + sub + max + cndmask + wmma, no EXEC churn;
// EXEC is all-ones at every WMMA (exact grid, uniform early-exit only).
// =====================================================================
__global__ void k_smb(const float* __restrict__ Zt, const float* __restrict__ H,
                      const float* __restrict__ Pd, const float* __restrict__ Td,
                      const float* __restrict__ mask, const float* __restrict__ mfp,
                      const int* __restrict__ needP, const int* __restrict__ yiP,
                      float* __restrict__ smb) {
  __shared__ float shT[368];
  __shared__ float shP[368];
  if (needP[0] == 0) return;  // uniform branch
  int yi = yiP[0];
  int tid = threadIdx.x;
  const float* Trow = Td + yi * NDAYS;
  const float* Prow = Pd + yi * NDAYS;
  for (int i = tid; i < NDAYS; i += 256) {
    unsigned ldsT = (unsigned)(uintptr_t)&shT[i];
    unsigned ldsP = (unsigned)(uintptr_t)&shP[i];
    unsigned off = (unsigned)(i * 4);
    asm volatile("global_load_async_to_lds_b32 %0, %1, %2"
                 :: "v"(ldsT), "v"(off), "s"((unsigned long long)(uintptr_t)Trow)
                 : "memory");
    asm volatile("global_load_async_to_lds_b32 %0, %1, %2"
                 :: "v"(ldsP), "v"(off), "s"((unsigned long long)(uintptr_t)Prow)
                 : "memory");
  }
  if (tid < 3) {                 // pad days 365..367: contribute exactly zero
    shT[NDAYS + tid] = -1.0e30f;
    shP[NDAYS + tid] = 0.0f;
  }
  asm volatile("s_wait_asynccnt 0" ::: "memory");
  __syncthreads();

  int lane = tid & 31;
  int wave = tid >> 5;
  int c = lane & 15;                          // A-matrix row = cell within wave
  int cell = blockIdx.x * 128 + wave * 16 + c;
  float a = 6.0e-3f * (Zt[cell] + H[cell] - 1500.0f);  // GAMMA * (Zs - Z_REF)

  v8f cpdd = {0.f, 0.f, 0.f, 0.f, 0.f, 0.f, 0.f, 0.f};
  v8f cacc = {0.f, 0.f, 0.f, 0.f, 0.f, 0.f, 0.f, 0.f};
  v2f bones = {1.0f, 1.0f};                   // B(4x16) = ones: D row-sum, layout-invariant
  int dbase = (lane >> 4) << 1;               // lanes 0-15 -> K 0/1, lanes 16-31 -> K 2/3

  for (int k0 = 0; k0 < 368; k0 += 4) {
    int d0 = k0 + dbase, d1 = d0 + 1;
    float T0 = shT[d0] - a;
    float T1 = shT[d1] - a;
    float p0 = shP[d0];
    float p1 = shP[d1];
    float r0 = fmaxf(T0, 0.f);
    float r1 = fmaxf(T1, 0.f);
    float s0 = (T0 <= 0.f) ? p0 : 0.f;        // cndmask, no branch
    float s1 = (T1 <= 0.f) ? p1 : 0.f;
    v2f ar = {r0, r1};
    v2f as = {s0, s1};
    cpdd = __builtin_amdgcn_wmma_f32_16x16x4_f32(false, ar, false, bones, (short)0, cpdd, false, false);
    cacc = __builtin_amdgcn_wmma_f32_16x16x4_f32(false, as, false, bones, (short)0, cacc, false, false);
  }

  // C/D layout: row m lives in component (m&7), lanes (m<8 ? 0-15 : 16-31); all N equal.
  float vp = mux8(cpdd, c & 7);
  float va = mux8(cacc, c & 7);
  int src = c + ((c >> 3) << 4);  // a lane in the correct half whose (lane&15)==c
  float pdd = __shfl(vp, src, 32);
  float acc = __shfl(va, src, 32);

  if (lane < 16) {
    float m = mask[cell];
    float mf = mfp[0];
    float s = (acc - mf * pdd) * m;
    smb[cell] = (s < 0.0f || m > 0.5f) ? s : -10.0f;
  }
}

// =====================================================================
__global__ void k_final(const float* __restrict__ Hfin, float* __restrict__ out) {
  int idx = blockIdx.x * 256 + threadIdx.x;
  out[3 * NTOT + idx] = Hfin[idx];
}

// =====================================================================
extern "C" void kernel_launch(void* const* d_in, const int* in_sizes, int n_in,
                              void* d_out, int out_size, void* d_ws, size_t ws_size,
                              hipStream_t stream) {
  (void)in_sizes; (void)n_in; (void)out_size; (void)ws_size;
  const float* Zt   = (const float*)d_in[0];
  const float* H0   = (const float*)d_in[1];
  const float* mask = (const float*)d_in[2];
  // d_in[3] precip_tensor, d_in[4] T_m_lowest, d_in[5] T_s: unused by reference
  const float* Pd   = (const float*)d_in[6];
  const float* Td   = (const float*)d_in[7];
  const float* mf   = (const float*)d_in[8];
  float* out = (float*)d_out;

  // workspace carve: 2x H ping-pong, D, smb, per-step maxD bits, scalar slots
  float* f = (float*)d_ws;
  float* Hb0  = f;
  float* Hb1  = f + NTOT;
  float* Dw   = f + 2 * NTOT;
  float* smbw = f + 3 * NTOT;
  unsigned* maxD = (unsigned*)(f + 4 * NTOT);
  float* ts = (float*)(maxD + NSTEPS);
  float* tl = ts + (NSTEPS + 1);
  int* pyi  = (int*)(tl + (NSTEPS + 1));
  int* g26  = pyi + (NSTEPS + 1);
  int* g57  = g26 + (NSTEPS + 1);
  int* g80  = g57 + (NSTEPS + 1);
  int* need = g80 + (NSTEPS + 1);
  int* yiA  = need + (NSTEPS + 1);

  const int GB = NTOT / 256;        // 1024 blocks of 256 (8 wave32s)
  const int SB = NTOT / 128;        // 2048 blocks: each wave owns 16 cells

  k_init<<<GB, 256, 0, stream>>>(H0, Hb0, out, maxD, ts, tl, pyi, g26, g57, g80, need, yiA);
  // initial SMB (unconditional slot NSTEPS, surface = Z_topo + H_init)
  k_smb<<<SB, 256, 0, stream>>>(Zt, Hb0, Pd, Td, mask, mf, need + NSTEPS, yiA + NSTEPS, smbw);

  float* buf[2] = {Hb0, Hb1};
  for (int i = 0; i < NSTEPS; ++i) {
    float* Hin = buf[i & 1];
    float* Hout = buf[(i + 1) & 1];
    k_diff<<<GB, 256, 0, stream>>>(Zt, Hin, Dw, maxD + i);
    k_update<<<GB, 256, 0, stream>>>(Zt, Hin, Hout, Dw, smbw, maxD,
                                     ts, tl, pyi, g26, g57, g80, need, yiA, i, out);
    k_smb<<<SB, 256, 0, stream>>>(Zt, Hout, Pd, Td, mask, mf, need + i, yiA + i, smbw);
  }
  k_final<<<GB, 256, 0, stream>>>(buf[NSTEPS & 1], out);
}